// SmoothnessPenalty_85469849190579
// MI455X (gfx1250) — compile-verified
//
#include <hip/hip_runtime.h>
#include <hip/hip_bf16.h>

#define KSEG 1024        // number of superpixel ids (matches reference K)
#define CG   8           // channels per channel-group block
#define BLK  256         // threads per block in the accumulation kernel
#define TILE (BLK * 4)   // pixels per staged tile (4 KB of int32 indices)

typedef __attribute__((ext_vector_type(2))) float v2f;
typedef __attribute__((ext_vector_type(8))) float v8f;

// Types matching __builtin_amdgcn_global_load_async_to_lds_b128:
// (v4i addrspace(1)* src, v4i addrspace(3)* dst, imm offset, imm cpol)
typedef int v4i __attribute__((vector_size(16)));
typedef __attribute__((address_space(1))) v4i gv4i;   // global (AS1)
typedef __attribute__((address_space(3))) v4i lv4i;   // LDS    (AS3)

#if __has_builtin(__builtin_amdgcn_global_load_async_to_lds_b128)
#define HAVE_ASYNC_LDS 1
#else
#define HAVE_ASYNC_LDS 0
#endif

// Stage 16 bytes per thread of the index array into LDS (async on gfx1250).
__device__ __forceinline__ void stage_idx_tile(const int* __restrict__ gsrc,
                                               int* __restrict__ ldst, int tid) {
#if HAVE_ASYNC_LDS
    __builtin_amdgcn_global_load_async_to_lds_b128(
        (gv4i*)(gsrc + (tid << 2)), (lv4i*)(ldst + (tid << 2)),
        /*imm offset=*/0, /*cpol=*/0);
#else
    *(int4*)(ldst + (tid << 2)) = *(const int4*)(gsrc + (tid << 2));
#endif
}

__device__ __forceinline__ void wait_async0() {
#if HAVE_ASYNC_LDS
#if __has_builtin(__builtin_amdgcn_s_wait_asynccnt)
    __builtin_amdgcn_s_wait_asynccnt(0);
#else
    asm volatile("s_wait_asynccnt 0" ::: "memory");
#endif
#endif
}

// ---------------------------------------------------------------------------
// Kernel 0: zero workspace (sums[D*K], counts[K], sumsq[1]) and the output.
// ---------------------------------------------------------------------------
__global__ void sp_zero_kernel(float* __restrict__ ws, int n, float* __restrict__ out) {
    int i = blockIdx.x * blockDim.x + threadIdx.x;
    if (i < n) ws[i] = 0.0f;
    if (i == 0) out[0] = 0.0f;
}

// ---------------------------------------------------------------------------
// Kernel 1: streaming segment accumulation.
//   grid.x = pixel chunks, grid.y = channel groups (D/CG).
//   LDS-privatized acc[CG][KSEG]; bank = k mod 64 -> conflict-free for
//   random segment ids. float4 (b128) coalesced embedding loads.
//   Superpixel indices are double-buffered in LDS via async global->LDS
//   copies (ASYNCcnt) so the consume phase never stalls on index returns.
// ---------------------------------------------------------------------------
__global__ void sp_seg_accum(const float* __restrict__ x,   // D x N embedding
                             const int*   __restrict__ sp,  // N segment ids
                             float* __restrict__ sums,      // D x K  (global)
                             float* __restrict__ counts,    // K      (global)
                             float* __restrict__ sumsq,     // 1      (global)
                             long long N) {
    __shared__ float acc[CG * KSEG];    // 32 KB
    __shared__ float cnt[KSEG];         // 4 KB (used by channel-group 0 only)
    __shared__ int   spbuf[2][TILE];    // 8 KB double-buffered index stage

    const int tid   = threadIdx.x;
    const int gbase = blockIdx.y * CG;

    for (int j = tid; j < CG * KSEG; j += BLK) acc[j] = 0.0f;
    if (blockIdx.y == 0)
        for (int j = tid; j < KSEG; j += BLK) cnt[j] = 0.0f;

    const long long per   = (N + gridDim.x - 1) / gridDim.x;
    const long long start = (long long)blockIdx.x * per;
    const long long end   = (start + per < N) ? (start + per) : N;

    float ss = 0.0f;

    const long long nfull = (end > start) ? ((end - start) / TILE) : 0;

    if (nfull > 0) {
        stage_idx_tile(sp + start, spbuf[0], tid);   // prologue stage of tile 0
        wait_async0();
    }
    __syncthreads();   // acc/cnt zeroed + tile 0 visible to all waves

    for (long long t = 0; t < nfull; ++t) {
        const long long tbase = start + t * TILE;
        const int cur = (int)(t & 1);

        // kick async stage of the next index tile + prefetch its embedding rows
        if (t + 1 < nfull) {
            stage_idx_tile(sp + tbase + TILE, spbuf[cur ^ 1], tid);
            const long long np = tbase + TILE + (tid << 2);
#pragma unroll
            for (int d = 0; d < CG; ++d)
                __builtin_prefetch(x + (long long)(gbase + d) * N + np);
        }

        // consume current tile: indices from LDS, embedding from global (b128)
        const long long p = tbase + (tid << 2);
        const int4 kk = *(const int4*)(&spbuf[cur][tid << 2]);
        const int k0 = kk.x & (KSEG - 1);
        const int k1 = kk.y & (KSEG - 1);
        const int k2 = kk.z & (KSEG - 1);
        const int k3 = kk.w & (KSEG - 1);
#pragma unroll
        for (int d = 0; d < CG; ++d) {
            const float4 v = *(const float4*)(x + (long long)(gbase + d) * N + p);
            ss += v.x * v.x + v.y * v.y + v.z * v.z + v.w * v.w;
            atomicAdd(&acc[d * KSEG + k0], v.x);   // ds_add_f32
            atomicAdd(&acc[d * KSEG + k1], v.y);
            atomicAdd(&acc[d * KSEG + k2], v.z);
            atomicAdd(&acc[d * KSEG + k3], v.w);
        }
        if (blockIdx.y == 0) {
            atomicAdd(&cnt[k0], 1.0f);
            atomicAdd(&cnt[k1], 1.0f);
            atomicAdd(&cnt[k2], 1.0f);
            atomicAdd(&cnt[k3], 1.0f);
        }

        wait_async0();     // my slice of the next tile has landed in LDS
        __syncthreads();   // everyone done reading cur + next tile complete
    }

    // scalar tail (< TILE pixels), indices straight from global
    for (long long p = start + nfull * TILE + tid; p < end; p += BLK) {
        const int k = sp[p] & (KSEG - 1);
#pragma unroll
        for (int d = 0; d < CG; ++d) {
            const float v = x[(long long)(gbase + d) * N + p];
            ss += v * v;
            atomicAdd(&acc[d * KSEG + k], v);
        }
        if (blockIdx.y == 0) atomicAdd(&cnt[k], 1.0f);
    }

    __syncthreads();

    // flush privatized bins: sums layout is [D][K] so local j maps flat
    for (int j = tid; j < CG * KSEG; j += BLK) {
        const float v = acc[j];
        if (v != 0.0f) unsafeAtomicAdd(&sums[(long long)gbase * KSEG + j], v);
    }
    if (blockIdx.y == 0) {
        for (int j = tid; j < KSEG; j += BLK) {
            const float v = cnt[j];
            if (v != 0.0f) unsafeAtomicAdd(&counts[j], v);
        }
    }

    // wave32 reduction of sum-of-squares, one global atomic per wave
#pragma unroll
    for (int off = 16; off > 0; off >>= 1) ss += __shfl_xor(ss, off, 32);
    if ((tid & 31) == 0) unsafeAtomicAdd(sumsq, ss);
}

// ---------------------------------------------------------------------------
// Kernel 2: finalize.  out = sumsq - sum_k (sum_d S[k][d]^2) / max(n_k, 1).
// Per 16-segment tile, sum_d S^2 is the diagonal of S*S^T computed on the
// matrix pipe with V_WMMA_F32_16X16X4_F32 (A and B fragments coincide by
// the ISA 16x4 / 4x16 f32 layouts).  One wave32 per tile.
// ---------------------------------------------------------------------------
__global__ void sp_finalize(const float* __restrict__ sums,    // D x K
                            const float* __restrict__ counts,  // K
                            const float* __restrict__ sumsq,   // 1
                            float* __restrict__ out,
                            int D) {
    const int lane = threadIdx.x;           // 32 lanes
    const int kb   = blockIdx.x * 16;       // segment tile base
    const int half = lane >> 4;             // 0: K-cols {0,1}; 1: K-cols {2,3}
    const int l    = lane & 15;             // matrix row within tile

    v8f cacc = {};
    for (int c = 0; c < D; c += 4) {
        const int d0 = c + half * 2;
        v2f a;
        a.x = sums[(long long)d0 * KSEG + kb + l];
        a.y = sums[(long long)(d0 + 1) * KSEG + kb + l];
        // D(16x16) += A(16x4) * B(4x16), B fragment == A fragment here
        cacc = __builtin_amdgcn_wmma_f32_16x16x4_f32(
            /*neg_a=*/false, a, /*neg_b=*/false, a,
            /*c_mod=*/(short)0, cacc, /*reuse_a=*/false, /*reuse_b=*/false);
    }

    // Diagonal of C per ISA layout: lanes 0-7 hold diag m=lane in c[lane];
    // lanes 24-31 hold diag m=lane-16 in c[lane-24].
    float part = 0.0f;
    if (lane < 8) {
        const float n = counts[kb + lane];
        part = cacc[lane] / fmaxf(n, 1.0f);
    } else if (lane >= 24) {
        const int m = lane - 16;
        const float n = counts[kb + m];
        part = cacc[lane - 24] / fmaxf(n, 1.0f);
    }

#pragma unroll
    for (int off = 16; off > 0; off >>= 1) part += __shfl_xor(part, off, 32);

    if (lane == 0) unsafeAtomicAdd(out, -part);
    if (blockIdx.x == 0 && lane == 0) unsafeAtomicAdd(out, sumsq[0]);
}

// ---------------------------------------------------------------------------
extern "C" void kernel_launch(void* const* d_in, const int* in_sizes, int n_in,
                              void* d_out, int out_size, void* d_ws, size_t ws_size,
                              hipStream_t stream) {
    const float* x  = (const float*)d_in[0];   // embedding, D*H*W f32
    const int*   sp = (const int*)d_in[1];     // superpixel ids, H*W
    const long long N = (long long)in_sizes[1];
    const int D = (int)((long long)in_sizes[0] / N);   // 64

    float* sums   = (float*)d_ws;                       // D*K
    float* counts = sums + (size_t)D * KSEG;            // K
    float* sumsq  = counts + KSEG;                      // 1
    float* out    = (float*)d_out;

    const int ztotal = D * KSEG + KSEG + 1;
    sp_zero_kernel<<<(ztotal + 255) / 256, 256, 0, stream>>>((float*)d_ws, ztotal, out);

    dim3 grid(32, (unsigned)(D / CG));                  // 32 pixel chunks x 8 ch groups
    sp_seg_accum<<<grid, BLK, 0, stream>>>(x, sp, sums, counts, sumsq, N);

    sp_finalize<<<KSEG / 16, 32, 0, stream>>>(sums, counts, sumsq, out, D);
}